// AmrstrongFrederickViscoplasticity_90975997264549
// MI455X (gfx1250) — compile-verified
//
#include <hip/hip_runtime.h>

// Armstrong–Frederick viscoplastic return mapping, thread-per-Gauss-point.
// Same 19-dim implicit Newton as the JAX reference (dp, depsp[6], da[12]),
// but each Newton step is solved by EXACT block elimination:
//   * analytic Jacobian blocks from N6 = dn/dsig_eff and colf = column scaling
//     of the constant -C@Dev operator,
//   * 6x6 solve for the back-stress coupling vector z (8 RHS at once),
//   * 7x7 Schur system in (dp, depsp),
//   * explicit recovery of the da blocks.
// All register-resident; LDS is only used for the CDNA5 async-staged eps tile.

namespace {

constexpr int   NEWTON    = 20;
constexpr float LAMc      = 115384.6153846154f;   // E*nu/((1+nu)(1-2nu))
constexpr float MUc       = 76923.076923076920f;  // E/(2(1+nu))
constexpr float LAM2MU    = LAMc + 2.0f * MUc;
constexpr float SIG0c     = 300.0f;
constexpr float SIGUc     = 500.0f;
constexpr float Bc        = 10.0f;
constexpr float Kc        = 500.0f;
constexpr float Gc        = 100.0f;
constexpr float CK0       = 40000.0f;
constexpr float CK1       = 4000.0f;
constexpr float DTc       = 0.01f;
constexpr float VM_FLOORc = 1e-8f * 200000.0f;    // 0.002
constexpr float TWO3      = 2.0f / 3.0f;
constexpr float SKc       = TWO3 * (CK0 + CK1);   // 29333.333

constexpr int BLK = 64;                            // 2 waves per block

} // namespace

__global__ __launch_bounds__(BLK)
void afvp_kernel(const float* __restrict__ eps,
                 const float* __restrict__ strain_old,
                 const float* __restrict__ p_old,
                 const float* __restrict__ epsp_old,
                 const float* __restrict__ a_old,
                 const float* __restrict__ sig_old,
                 float* __restrict__ out, int N)
{
    extern __shared__ float St[];                 // BLK*6 staged eps
    const int tid = threadIdx.x;
    const int g   = blockIdx.x * BLK + tid;

    // ---- CDNA5 async global->LDS staging of the eps tile (gfx1250 path) ----
    float deps[6], sigo[6], epso[6], a0c[12];
#if defined(__AMDGCN__) && __has_builtin(__builtin_amdgcn_global_load_async_to_lds_b32)
    {
        typedef __attribute__((address_space(1))) int gint_t;
        typedef __attribute__((address_space(3))) int lint_t;
        const float* gbase = eps + (size_t)blockIdx.x * BLK * 6;
        #pragma unroll
        for (int c = 0; c < 6; ++c) {
            gint_t* gp = (gint_t*)(gbase + c * BLK + tid);
            lint_t* lp = (lint_t*)(St + c * BLK + tid);
            __builtin_amdgcn_global_load_async_to_lds_b32(gp, lp, 0, 0);
        }
#if __has_builtin(__builtin_amdgcn_s_wait_asynccnt)
        __builtin_amdgcn_s_wait_asynccnt(0);
#else
        asm volatile("s_wait_asynccnt 0" ::: "memory");
#endif
        asm volatile("" ::: "memory");
    }
    __syncthreads();
    if (g >= N) return;
    #pragma unroll
    for (int c = 0; c < 6; ++c)
        deps[c] = St[tid * 6 + c] - strain_old[(size_t)g * 6 + c];
#else
    if (g >= N) return;
    #pragma unroll
    for (int c = 0; c < 6; ++c)
        deps[c] = eps[(size_t)g * 6 + c] - strain_old[(size_t)g * 6 + c];
#endif

    #pragma unroll
    for (int c = 0; c < 6; ++c) {
        sigo[c] = sig_old[(size_t)g * 6 + c];
        epso[c] = epsp_old[(size_t)g * 6 + c];
    }
    #pragma unroll
    for (int k = 0; k < 12; ++k) a0c[k] = a_old[(size_t)g * 12 + k];
    const float p0 = p_old[g];

    // Unknowns: dp, e[6]=depsp, d1[6], d2[6] = da
    float dp = 0.0f, e[6], d1[6], d2[6];
    #pragma unroll
    for (int c = 0; c < 6; ++c) { e[c] = 0.0f; d1[c] = 0.0f; d2[c] = 0.0f; }

    for (int it = 0; it < NEWTON; ++it) {
        // ================= value pass =================
        const float trv = (e[0] + e[1] + e[2]) * (1.0f / 3.0f);
        float w[6];
        #pragma unroll
        for (int c = 0; c < 3; ++c) w[c] = deps[c] - (e[c] - trv);
        #pragma unroll
        for (int c = 3; c < 6; ++c) w[c] = deps[c] - e[c];

        float am1[6], am2[6], se[6];
        #pragma unroll
        for (int c = 0; c < 6; ++c) { am1[c] = a0c[c] + d1[c]; am2[c] = a0c[6 + c] + d2[c]; }
        #pragma unroll
        for (int c = 0; c < 3; ++c) {
            const int c1 = (c + 1) % 3, c2 = (c + 2) % 3;
            se[c] = sigo[c] + LAM2MU * w[c] + LAMc * (w[c1] + w[c2])
                  - TWO3 * (CK0 * am1[c] + CK1 * am2[c]);
        }
        #pragma unroll
        for (int c = 3; c < 6; ++c)
            se[c] = sigo[c] + MUc * w[c] - TWO3 * (CK0 * am1[c] + CK1 * am2[c]);

        const float tev = (se[0] + se[1] + se[2]) * (1.0f / 3.0f);
        float s[6];
        #pragma unroll
        for (int c = 0; c < 3; ++c) s[c] = se[c] - tev;
        #pragma unroll
        for (int c = 3; c < 6; ++c) s[c] = se[c];

        const float j2 = 0.5f * (s[0] * s[0] + s[1] * s[1] + s[2] * s[2])
                       + s[3] * s[3] + s[4] * s[4] + s[5] * s[5];
        const float vm  = sqrtf(3.0f * j2 + 1e-16f);
        const float ivm = 1.0f / vm;
        float n[6];
        #pragma unroll
        for (int c = 0; c < 6; ++c) n[c] = ((c < 3) ? 1.5f : 3.0f) * s[c] * ivm;

        const bool  gate = (vm > VM_FLOORc);
        const float ev    = expf(-Bc * (p0 + dp));
        const float voce  = SIG0c + (SIGUc - SIG0c) * (1.0f - ev);
        const float vocep = (SIGUc - SIG0c) * Bc * ev;
        const float f     = (gate ? vm : VM_FLOORc) - voce;

        float nor = 0.0f, norp = 0.0f;
        if (f > 0.0f) {
            const float q = f * (1.0f / Kc);
            nor  = q * q * q;
            norp = 3.0f * q * q * (1.0f / Kc);
        }
        const float dn  = DTc * norp;          // DT * nor'
        const float gdn = gate ? dn : 0.0f;

        // residual values
        const float r1 = dp - DTc * nor;
        float r2[6], r31[6], r32[6];
        #pragma unroll
        for (int c = 0; c < 6; ++c) {
            r2[c]  = e[c] - n[c] * dp;
            r31[c] = d1[c] - dp * (n[c] - Gc * am1[c]);
            r32[c] = d2[c] - dp * (n[c] - Gc * am2[c]);
        }

        // ================= analytic Jacobian blocks =================
        // N6 = dn/dse = ivm*(diag(fct)*P - n n^T)
        float N6[6][6];
        #pragma unroll
        for (int c = 0; c < 6; ++c) {
            const float fct = (c < 3) ? 1.5f : 3.0f;
            #pragma unroll
            for (int d = 0; d < 6; ++d) {
                float P;
                if (c < 3 && d < 3)      P = ((c == d) ? 1.0f : 0.0f) - (1.0f / 3.0f);
                else if (c >= 3 && d >= 3) P = (c == d) ? 1.0f : 0.0f;
                else                      P = 0.0f;
                N6[c][d] = ivm * (fct * P - n[c] * n[d]);
            }
        }
        // A_cd = colf[d]*N6[c][d] = (N6 * dse/de)_cd ;  u_d = colf[d]*n[d] = (n^T dse/de)_d
        float colf[6];
        #pragma unroll
        for (int d = 0; d < 6; ++d) colf[d] = (d < 3) ? (-2.0f * MUc) : (-MUc);

        // ===== 6x6 solve: M = (1+G dp)I + dp*SK*N6 ; RHS = [bz | gz | dp*SK*A] =====
        const float dpSK = dp * SKc;
        float W[6][14];
        #pragma unroll
        for (int c = 0; c < 6; ++c) {
            #pragma unroll
            for (int d = 0; d < 6; ++d)
                W[c][d] = ((c == d) ? (1.0f + Gc * dp) : 0.0f) + dpSK * N6[c][d];
            W[c][6] = TWO3 * (CK0 * r31[c] + CK1 * r32[c]);                       // bz
            W[c][7] = SKc * n[c] - TWO3 * Gc * (CK0 * am1[c] + CK1 * am2[c]);     // gz
            #pragma unroll
            for (int d = 0; d < 6; ++d)
                W[c][8 + d] = dpSK * colf[d] * N6[c][d];                          // dp*SK*A
        }
        // Gaussian elimination (no pivot; M ~ I) + back-substitution for 8 RHS
        #pragma unroll
        for (int k = 0; k < 6; ++k) {
            const float inv = 1.0f / W[k][k];
            #pragma unroll
            for (int i = k + 1; i < 6; ++i) {
                const float m = W[i][k] * inv;
                #pragma unroll
                for (int j = k + 1; j < 14; ++j)
                    W[i][j] = fmaf(-m, W[k][j], W[i][j]);
            }
        }
        #pragma unroll
        for (int i = 5; i >= 0; --i) {
            const float inv = 1.0f / W[i][i];
            #pragma unroll
            for (int q = 6; q < 14; ++q) {
                float acc = W[i][q];
                #pragma unroll
                for (int j = i + 1; j < 6; ++j)
                    acc = fmaf(-W[i][j], W[j][q], acc);
                W[i][q] = acc * inv;
            }
        }
        // Now: zb = W[:,6], zg = W[:,7], Ze = W[:,8+d]

        // ===== 7x7 Schur system in (x0, xe) =====
        float Q[7][8];
        {
            float ngz = 0.0f, ngb = 0.0f;      // n^T zg, n^T zb
            #pragma unroll
            for (int c = 0; c < 6; ++c) { ngz += n[c] * W[c][7]; ngb += n[c] * W[c][6]; }
            Q[0][0] = 1.0f + dn * vocep + gdn * ngz;                  // alpha
            Q[0][7] = r1 - gdn * ngb;                                 // rho
            #pragma unroll
            for (int d = 0; d < 6; ++d) {
                float nZ = 0.0f;               // (n^T Ze)_d
                #pragma unroll
                for (int c = 0; c < 6; ++c) nZ += n[c] * W[c][8 + d];
                Q[0][1 + d] = -gdn * colf[d] * n[d] + gdn * nZ;       // v_d
            }
            #pragma unroll
            for (int c = 0; c < 6; ++c) {
                float nzg = 0.0f, nzb = 0.0f;  // (N6 zg)_c, (N6 zb)_c
                #pragma unroll
                for (int ee = 0; ee < 6; ++ee) {
                    nzg = fmaf(N6[c][ee], W[ee][7], nzg);
                    nzb = fmaf(N6[c][ee], W[ee][6], nzb);
                }
                Q[1 + c][0] = -n[c] + dp * nzg;                        // c_vec
                Q[1 + c][7] = r2[c] - dp * nzb;                        // rhs_e
                #pragma unroll
                for (int d = 0; d < 6; ++d) {
                    float nz = 0.0f;           // (N6 Ze)_cd
                    #pragma unroll
                    for (int ee = 0; ee < 6; ++ee)
                        nz = fmaf(N6[c][ee], W[ee][8 + d], nz);
                    Q[1 + c][1 + d] = ((c == d) ? 1.0f : 0.0f)
                                    - dp * colf[d] * N6[c][d] + dp * nz;   // B
                }
            }
        }
        #pragma unroll
        for (int k = 0; k < 7; ++k) {
            const float inv = 1.0f / Q[k][k];
            #pragma unroll
            for (int i = k + 1; i < 7; ++i) {
                const float m = Q[i][k] * inv;
                #pragma unroll
                for (int j = k + 1; j < 8; ++j)
                    Q[i][j] = fmaf(-m, Q[k][j], Q[i][j]);
            }
        }
        #pragma unroll
        for (int i = 6; i >= 0; --i) {
            float acc = Q[i][7];
            #pragma unroll
            for (int j = i + 1; j < 7; ++j)
                acc = fmaf(-Q[i][j], Q[j][7], acc);
            Q[i][7] = acc / Q[i][i];
        }
        const float x0 = Q[0][7];
        float xe[6];
        #pragma unroll
        for (int d = 0; d < 6; ++d) xe[d] = Q[1 + d][7];

        // ===== recover z, then xa1 / xa2, then Newton update =====
        float z[6];
        #pragma unroll
        for (int c = 0; c < 6; ++c) {
            float acc = W[c][6] + x0 * W[c][7];
            #pragma unroll
            for (int d = 0; d < 6; ++d) acc = fmaf(W[c][8 + d], xe[d], acc);
            z[c] = acc;
        }
        const float inv1 = 1.0f / (1.0f + Gc * dp);
        #pragma unroll
        for (int c = 0; c < 6; ++c) {
            float Axe = 0.0f, N6z = 0.0f;
            #pragma unroll
            for (int d = 0; d < 6; ++d) {
                Axe = fmaf(colf[d] * N6[c][d], xe[d], Axe);
                N6z = fmaf(N6[c][d], z[d], N6z);
            }
            const float x1 = (r31[c] + (n[c] - Gc * am1[c]) * x0 + dp * Axe - dp * N6z) * inv1;
            const float x2 = (r32[c] + (n[c] - Gc * am2[c]) * x0 + dp * Axe - dp * N6z) * inv1;
            d1[c] -= x1;
            d2[c] -= x2;
            e[c]  -= xe[c];
        }
        dp -= x0;
    }

    // ---------- outputs: sig, p, epsp, a  (concat flat, return order) ----------
    {
        const float trp = (e[0] + e[1] + e[2]) * (1.0f / 3.0f);
        float wv[6];
        #pragma unroll
        for (int c = 0; c < 3; ++c) wv[c] = deps[c] - (e[c] - trp);
        #pragma unroll
        for (int c = 3; c < 6; ++c) wv[c] = deps[c] - e[c];
        float sg[6];
        #pragma unroll
        for (int c = 0; c < 3; ++c) {
            const int c1 = (c + 1) % 3, c2 = (c + 2) % 3;
            sg[c] = sigo[c] + LAM2MU * wv[c] + LAMc * (wv[c1] + wv[c2]);
        }
        #pragma unroll
        for (int c = 3; c < 6; ++c) sg[c] = sigo[c] + MUc * wv[c];

        const size_t b6 = (size_t)g * 6;
        #pragma unroll
        for (int c = 0; c < 6; ++c) out[b6 + c] = sg[c];
        out[(size_t)N * 6 + g] = p0 + dp;
        #pragma unroll
        for (int c = 0; c < 6; ++c) out[(size_t)N * 7 + b6 + c] = epso[c] + e[c];
        const size_t b12 = (size_t)g * 12;
        #pragma unroll
        for (int c = 0; c < 6; ++c) {
            out[(size_t)N * 13 + b12 + c]     = a0c[c]     + d1[c];
            out[(size_t)N * 13 + b12 + 6 + c] = a0c[6 + c] + d2[c];
        }
    }
}

extern "C" void kernel_launch(void* const* d_in, const int* in_sizes, int n_in,
                              void* d_out, int out_size, void* d_ws, size_t ws_size,
                              hipStream_t stream)
{
    const float* eps        = (const float*)d_in[0];
    const float* strain_old = (const float*)d_in[1];
    const float* p_old      = (const float*)d_in[2];
    const float* epsp_old   = (const float*)d_in[3];
    const float* a_old      = (const float*)d_in[4];
    const float* sig_old    = (const float*)d_in[5];
    float* out = (float*)d_out;

    const int N = in_sizes[2];                    // p_old has N elements
    const int blocks = (N + BLK - 1) / BLK;
    const size_t shmem = (size_t)BLK * 6 * sizeof(float);   // staged eps tile

    afvp_kernel<<<blocks, BLK, shmem, stream>>>(eps, strain_old, p_old, epsp_old,
                                                a_old, sig_old, out, N);
}